// NDP_1726576855107
// MI455X (gfx1250) — compile-verified
//
#include <hip/hip_runtime.h>
#include <math.h>

// CDNA5 / gfx1250, wave32.
typedef __attribute__((ext_vector_type(2))) float v2f;
typedef __attribute__((ext_vector_type(8))) float v8f;

// ---------------------------------------------------------------------------
// C[M,Ncols] = act(A[M,K] @ B[K,Ncols] + bias) using V_WMMA_F32_16X16X4_F32.
// grid.x = M/16, block = 32 * (Ncols/16) threads (Ncols=128 -> 256 threads).
// Fragment layouts per CDNA5 ISA §7.12.2:
//   A 16x4 (2 VGPRs): lane l -> m = l%16, kg = (l/16)*2; v0=K[kg], v1=K[kg+1]
//   B  4x16 (2 VGPRs): lane l -> n = l%16, kg = (l/16)*2 (mirror of A)
//   C/D 16x16 (8 VGPRs): reg r -> row = r + (l>=16 ? 8 : 0), col = l%16
// ---------------------------------------------------------------------------
template<bool RELU, bool BIAS>
__global__ __launch_bounds__(256)
void wmma_gemm_f32(const float* __restrict__ A, const float* __restrict__ B,
                   const float* __restrict__ bias, float* __restrict__ C,
                   int M, int Ncols, int K)
{
  const int lane = threadIdx.x & 31;
  const int wave = threadIdx.x >> 5;
  const int row0 = blockIdx.x << 4;
  const int col0 = wave << 4;
  const int mn   = lane & 15;          // m for A frag / n for B frag & C
  const int kg   = (lane >> 4) << 1;   // K sub-offset within group-of-4

  v8f acc = {0.f, 0.f, 0.f, 0.f, 0.f, 0.f, 0.f, 0.f};
  const float* Ar = A + (size_t)(row0 + mn) * K + kg;        // row of A
  const float* Bc = B + (size_t)kg * Ncols + col0 + mn;       // col of B

  for (int kk = 0; kk < K; kk += 4) {
    v2f a, b;
    a.x = Ar[kk + 0];
    a.y = Ar[kk + 1];
    b.x = Bc[(size_t)(kk + 0) * Ncols];
    b.y = Bc[(size_t)(kk + 1) * Ncols];
    acc = __builtin_amdgcn_wmma_f32_16x16x4_f32(
        /*neg_a=*/false, a, /*neg_b=*/false, b,
        /*c_mod=*/(short)0, acc, /*reuse_a=*/false, /*reuse_b=*/false);
  }

  const float bv = BIAS ? bias[col0 + mn] : 0.0f;
  const int rbase = row0 + ((lane >> 4) << 3);
#pragma unroll
  for (int r = 0; r < 8; ++r) {
    float v = acc[r] + bv;
    if (RELU) v = fmaxf(v, 0.0f);
    C[(size_t)(rbase + r) * Ncols + col0 + mn] = v;
  }
}

// ---------------------------------------------------------------------------
// repl[row] = sigmoid(dot(S1[row,:128], R2[:128]) + rb2). One wave per row.
// ---------------------------------------------------------------------------
__global__ __launch_bounds__(256)
void repl_kernel(const float* __restrict__ S1, const float* __restrict__ R2,
                 const float* __restrict__ rb2, float* __restrict__ out)
{
  const int lane = threadIdx.x & 31;
  const int row  = blockIdx.x * 8 + (threadIdx.x >> 5);
  float s = 0.f;
#pragma unroll
  for (int k = 0; k < 4; ++k)
    s += S1[(size_t)row * 128 + lane + k * 32] * R2[lane + k * 32];
#pragma unroll
  for (int off = 16; off > 0; off >>= 1)
    s += __shfl_xor(s, off, 32);
  if (lane == 0)
    out[row] = 1.0f / (1.0f + expf(-(s + rb2[0])));
}

// ---------------------------------------------------------------------------
// edge[i,j] = sum_k relu(Hi[i,k] + Hj[j,k]) * U2[k] + ub2   (ub1 folded in Hi)
// ReLU between broadcast-add and dot => not a GEMM; register-blocked VALU.
// 64(i) x 32(j) LDS tile per 256-thread block; 4x2 outputs per thread.
// sHi reads are half-wave broadcasts (no conflicts); sHj padded to 129 so the
// 16 j-groups map to 16 distinct LDS banks (stride 2*129 % 64 per group).
// ---------------------------------------------------------------------------
__global__ __launch_bounds__(256)
void edge_kernel(const float* __restrict__ Hi, const float* __restrict__ Hj,
                 const float* __restrict__ U2, const float* __restrict__ ub2,
                 float* __restrict__ E)
{
  __shared__ float sHi[64][128];
  __shared__ float sHj[32][129];
  __shared__ float sU2[128];

  const int t  = threadIdx.x;
  const int ib = blockIdx.x * 64;
  const int jb = blockIdx.y * 32;

  // Fill Hi tile: 64x128 = 2048 float4, 8 per thread (coalesced b128 loads)
#pragma unroll
  for (int it = 0; it < 8; ++it) {
    int idx = t + it * 256;
    int r = idx >> 5, c = (idx & 31) << 2;
    const float4 v = *(const float4*)(Hi + (size_t)(ib + r) * 128 + c);
    sHi[r][c+0] = v.x; sHi[r][c+1] = v.y; sHi[r][c+2] = v.z; sHi[r][c+3] = v.w;
  }
  // Fill Hj tile: 32x128 = 1024 float4, 4 per thread
#pragma unroll
  for (int it = 0; it < 4; ++it) {
    int idx = t + it * 256;
    int r = idx >> 5, c = (idx & 31) << 2;
    const float4 v = *(const float4*)(Hj + (size_t)(jb + r) * 128 + c);
    sHj[r][c+0] = v.x; sHj[r][c+1] = v.y; sHj[r][c+2] = v.z; sHj[r][c+3] = v.w;
  }
  if (t < 128) sU2[t] = U2[t];
  __syncthreads();

  const float c2 = ub2[0];
  const int i0 = (t >> 4) << 2;   // 0,4,...,60   (constant per half-wave)
  const int j0 = (t & 15) << 1;   // 0,2,...,30

  float acc[4][2] = {};
#pragma unroll 4
  for (int k = 0; k < 128; ++k) {
    const float ck = sU2[k];
    const float b0 = sHj[j0 + 0][k];
    const float b1 = sHj[j0 + 1][k];
#pragma unroll
    for (int ii = 0; ii < 4; ++ii) {
      const float a = sHi[i0 + ii][k];
      acc[ii][0] += fmaxf(a + b0, 0.f) * ck;
      acc[ii][1] += fmaxf(a + b1, 0.f) * ck;
    }
  }
#pragma unroll
  for (int ii = 0; ii < 4; ++ii) {
    float2 o = make_float2(acc[ii][0] + c2, acc[ii][1] + c2);
    *(float2*)(E + (size_t)(ib + i0 + ii) * 1024 + jb + j0) = o;
  }
}

// ---------------------------------------------------------------------------
extern "C" void kernel_launch(void* const* d_in, const int* in_sizes, int n_in,
                              void* d_out, int out_size, void* d_ws, size_t ws_size,
                              hipStream_t stream)
{
  const float* X   = (const float*)d_in[0];   // node_embeddings [1024,128]
  // d_in[1] adjacency_matrix: unused by the reference computation
  const float* W1  = (const float*)d_in[2];   // [128,128]
  const float* b1  = (const float*)d_in[3];   // [128]
  const float* W2  = (const float*)d_in[4];   // [128,128]
  const float* b2  = (const float*)d_in[5];   // [128]
  const float* R1  = (const float*)d_in[6];   // [128,128]
  const float* rb1 = (const float*)d_in[7];   // [128]
  const float* R2  = (const float*)d_in[8];   // [128,1]
  const float* rb2 = (const float*)d_in[9];   // [1]
  const float* U1  = (const float*)d_in[10];  // [256,128]
  const float* ub1 = (const float*)d_in[11];  // [128]
  const float* U2  = (const float*)d_in[12];  // [128,1]
  const float* ub2 = (const float*)d_in[13];  // [1]

  const int N = 1024, D = 128, H = 128;

  float* out  = (float*)d_out;
  float* u    = out;                       // [1024,128]
  float* repl = out + (size_t)N * D;       // [1024]
  float* E    = repl + N;                  // [1024,1024]

  float* ws = (float*)d_ws;
  float* T1 = ws;                          // [1024,128] tmp; reused as S1
  float* HI = ws + 131072;                 // [1024,128] u@U1[:D] + ub1
  float* HJ = ws + 262144;                 // [1024,128] u@U1[D:]

  dim3 blk(256);
  // u = relu(X@W1+b1)@W2 + b2
  wmma_gemm_f32<true , true ><<<N/16, blk, 0, stream>>>(X,  W1, b1,  T1, N, H, D);
  wmma_gemm_f32<false, true ><<<N/16, blk, 0, stream>>>(T1, W2, b2,  u,  N, D, H);
  // repl = sigmoid(relu(u@R1+rb1) @ R2 + rb2)
  wmma_gemm_f32<true , true ><<<N/16, blk, 0, stream>>>(u,  R1, rb1, T1, N, H, D);
  repl_kernel<<<N/8, blk, 0, stream>>>(T1, R2, rb2, repl);
  // hi = u@U1[:D] + ub1 (bias folded), hj = u@U1[D:]
  wmma_gemm_f32<false, true ><<<N/16, blk, 0, stream>>>(u, U1,                ub1,     HI, N, H, D);
  wmma_gemm_f32<false, false><<<N/16, blk, 0, stream>>>(u, U1 + (size_t)D*H,  nullptr, HJ, N, H, D);
  // edge[i,j] = relu(hi_i + hj_j) . U2 + ub2
  edge_kernel<<<dim3(N/64, N/32), blk, 0, stream>>>(HI, HJ, U2, ub2, E);
}